// Conv2d_NN_1494648619730
// MI455X (gfx1250) — compile-verified
//
#include <hip/hip_runtime.h>
#include <hip/hip_bf16.h>

// Problem constants (from reference): x [8,32,64,64], weight [32,32,9], bias [32]
#define BB   8
#define CCH  32
#define TT   4096     // 64*64 tokens
#define KNN  9
#define OCH  32

typedef __attribute__((ext_vector_type(2))) float v2f;
typedef __attribute__((ext_vector_type(4))) float v4f;
typedef __attribute__((ext_vector_type(8))) float v8f;

// Compiler-only memory ordering: the sim/topk workgroup is a single wave and
// LDS ops from one wave complete in program order (DScnt in-order), so no
// hardware barrier is needed -- only a compiler reordering fence, since the
// cross-lane LDS exchange uses different per-thread store/load addresses.
#define WAVE_LDS_FENCE() asm volatile("" ::: "memory")

// ---------------------------------------------------------------------------
// Kernel 1: channel-normalize each token; write xn token-major [B][T][C]
// so WMMA A/B fragment loads are contiguous float2 per lane.
// ---------------------------------------------------------------------------
__global__ __launch_bounds__(256) void normalize_kernel(const float* __restrict__ x,
                                                        float* __restrict__ xn) {
    int gid = blockIdx.x * blockDim.x + threadIdx.x;   // token over B*T
    if (gid >= BB * TT) return;
    int b = gid >> 12;            // / 4096
    int t = gid & (TT - 1);
    const float* xb = x + (size_t)b * CCH * TT + t;    // x[b][c][t], stride T in c
    float v[CCH];
    float s = 0.0f;
#pragma unroll
    for (int c = 0; c < CCH; ++c) {
        v[c] = xb[(size_t)c * TT];
        s += v[c] * v[c];
    }
    float inv = 1.0f / fmaxf(sqrtf(s), 1e-12f);        // F.normalize semantics
    float* o = xn + (size_t)gid * CCH;
#pragma unroll
    for (int c = 0; c < CCH; ++c) o[c] = v[c] * inv;
}

// ---------------------------------------------------------------------------
// Kernel 2: fused similarity GEMM (fp32 WMMA 16x16x4, 8 chained per tile)
// + clip + streaming top-9 per row. One wave per 16-row t-tile.
// Software-pipelined: next tile's B fragments are issued before the top-k
// scan, and no counter-draining barrier intervenes, so the scan fully hides
// the load latency. Top-9 insert chain skipped via wave-uniform ballot.
// ---------------------------------------------------------------------------
#define SROW 20   // sim tile row stride (floats): 80B => 16B-aligned rows,
                  // and r*20 mod 64 distinct for r=0..15 (conflict-free)

__global__ __launch_bounds__(32) void simtopk_kernel(const float* __restrict__ xn,
                                                     float* __restrict__ topv,
                                                     int* __restrict__ topi) {
    __shared__ float sims[16 * SROW];    // 16x16 sim tile, padded row stride
    __shared__ float mV[32 * KNN];       // per-lane partial top-9 values
    __shared__ int   mI[32 * KNN];       // per-lane partial top-9 indices

    const int wg   = blockIdx.x;         // b*256 + t-tile
    const int b    = wg >> 8;
    const int t0   = (wg & 255) << 4;
    const int lane = threadIdx.x;
    const int half = lane >> 4;          // 0: lanes 0-15, 1: lanes 16-31
    const int l15  = lane & 15;
    const int koff = half * 2;           // K-pair offset per ISA 16x4 f32 layout

    // A fragments: A[m][k] = xn[b][t0+m][k]; lane m=l15, (k, k+1) = kc*4+koff
    v2f a[8];
    const float* arow = xn + ((size_t)(b * TT + t0 + l15)) * CCH;
#pragma unroll
    for (int kc = 0; kc < 8; ++kc)
        a[kc] = *(const v2f*)(arow + kc * 4 + koff);

    // Per-lane base for B fragments; tile st adds st*16*CCH floats
    const float* xnb = xn + (size_t)b * TT * CCH + (size_t)l15 * CCH + koff;

    // Streaming top-9 (descending, strict-> insert => lowest-index tie-break)
    float tv[KNN]; int ti[KNN];
#pragma unroll
    for (int k = 0; k < KNN; ++k) { tv[k] = -2.0f; ti[k] = 0; }

    v2f bf0[8], bf1[8];
#pragma unroll
    for (int kc = 0; kc < 8; ++kc)                 // prologue: tile 0 fragments
        bf0[kc] = *(const v2f*)(xnb + kc * 4);

    // One pipelined tile step: WMMA over bc, prefetch tile pf into bn,
    // then clip/store/top-k for tile st while the prefetch is in flight.
    auto tile_step = [&](v2f (&bc)[8], v2f (&bn)[8], int st, int pf) {
        const int s0 = st << 4;
        v8f acc = {0.f, 0.f, 0.f, 0.f, 0.f, 0.f, 0.f, 0.f};
#pragma unroll
        for (int kc = 0; kc < 8; ++kc)
            acc = __builtin_amdgcn_wmma_f32_16x16x4_f32(
                false, a[kc], false, bc[kc], (short)0, acc, false, false);

        // issue next tile's B-fragment loads (consumed by the NEXT tile_step)
        const float* nb = xnb + (size_t)pf * (16 * CCH);
#pragma unroll
        for (int kc = 0; kc < 8; ++kc)
            bn[kc] = *(const v2f*)(nb + kc * 4);

        // clip to [-1,1]; D layout: lane holds rows m = vg + half*8, col n = l15
#pragma unroll
        for (int vg = 0; vg < 8; ++vg) {
            float f = fminf(fmaxf(acc[vg], -1.0f), 1.0f);
            sims[(vg + half * 8) * SROW + l15] = f;
        }
        WAVE_LDS_FENCE();                // order DS store -> DS load (same wave)
        // lane (half,l15): row r=l15 owns candidates n = half*8 + [0..7]
        const v4f* crow = (const v4f*)&sims[l15 * SROW + half * 8];
        v4f c0 = crow[0];
        v4f c1 = crow[1];
        WAVE_LDS_FENCE();                // order DS load -> next tile's DS store
#pragma unroll
        for (int j = 0; j < 8; ++j) {
            float cv = (j < 4) ? c0[j & 3] : c1[j & 3];
            int   ci = s0 + half * 8 + j;
            // wave-uniform skip: insert chain is a no-op unless cv > current min
            unsigned anyIns = __builtin_amdgcn_ballot_w32(cv > tv[KNN - 1]);
            if (anyIns) {
#pragma unroll
                for (int k = 0; k < KNN; ++k) {     // register bubble-insert
                    bool  gt = cv > tv[k];
                    float pv = tv[k]; int pi = ti[k];
                    tv[k] = gt ? cv : pv;
                    ti[k] = gt ? ci : pi;
                    cv    = gt ? pv : cv;
                    ci    = gt ? pi : ci;
                }
            }
        }
    };

    for (int st = 0; st < 256; st += 2) {
        int pf1 = st + 1;                               // always <= 255
        tile_step(bf0, bf1, st, pf1);
        int pf2 = (st + 2 < 256) ? (st + 2) : 255;      // clamped (redundant, in-bounds)
        tile_step(bf1, bf0, st + 1, pf2);
    }

    // Merge the two half-lane lists for each row (value desc, index asc on ties)
#pragma unroll
    for (int k = 0; k < KNN; ++k) { mV[lane * KNN + k] = tv[k]; mI[lane * KNN + k] = ti[k]; }
    WAVE_LDS_FENCE();                    // same-wave cross-lane exchange
    if (lane < 16) {
        const float* va = &mV[lane * KNN];        const int* ia = &mI[lane * KNN];
        const float* vb = &mV[(lane + 16) * KNN]; const int* ib = &mI[(lane + 16) * KNN];
        float* ov = topv + ((size_t)(b * TT + t0 + lane)) * KNN;
        int*   oi = topi + ((size_t)(b * TT + t0 + lane)) * KNN;
        int i = 0, j = 0;
#pragma unroll
        for (int k = 0; k < KNN; ++k) {
            float fa = va[i], fb = vb[j];
            int   xa = ia[i], xb = ib[j];
            bool pickA = (fa > fb) || ((fa == fb) && (xa < xb));
            ov[k] = pickA ? fa : fb;
            oi[k] = pickA ? xa : xb;
            i += pickA ? 1 : 0;
            j += pickA ? 0 : 1;
        }
    }
}

// ---------------------------------------------------------------------------
// Kernel 3: gather neighbors, scale by similarity, contract over (C,K)=288
// with weight. Weight + gathered tile staged in LDS (≈55 KB of 320 KB/WGP);
// inner contraction uses float4 LDS loads on both operands.
// ---------------------------------------------------------------------------
#define TOK 16
#define PRPAD 292   // 288 padded (multiple of 4 floats => 16B rows; 16 row
                    // starts hit distinct LDS banks)

__global__ __launch_bounds__(512) void gatherconv_kernel(const float* __restrict__ x,
                                                         const float* __restrict__ weight,
                                                         const float* __restrict__ bias,
                                                         const float* __restrict__ topv,
                                                         const int* __restrict__ topi,
                                                         float* __restrict__ out) {
    __shared__ float wsh[OCH * 288];    // weight[o][c*9+k], contiguous per o
    __shared__ float pr[TOK * PRPAD];   // prime[token][c*9+k]

    const int b   = blockIdx.x >> 8;
    const int t0  = (blockIdx.x & 255) << 4;
    const int tid = threadIdx.x;

    for (int i = tid; i < OCH * 288; i += 512) wsh[i] = weight[i];

    const float* xb = x + (size_t)b * CCH * TT;
    for (int e = tid; e < TOK * 288; e += 512) {
        int tt = e / 288;  int cc = e - tt * 288;
        int c  = cc / KNN; int kk = cc - c * KNN;
        size_t tb = ((size_t)(b * TT + t0 + tt)) * KNN + kk;
        float v    = topv[tb];
        int   sidx = topi[tb];
        pr[tt * PRPAD + cc] = v * xb[(size_t)c * TT + sidx];
    }
    __syncthreads();                     // real cross-wave barrier (16 waves)

    int tt = tid & 15;
    int o  = tid >> 4;
    float acc = bias[o];
    const v4f* wrow = (const v4f*)&wsh[o * 288];     // 288*4B row: 16B-aligned
    const v4f* prow = (const v4f*)&pr[tt * PRPAD];   // 292*4B row: 16B-aligned
#pragma unroll 8
    for (int q = 0; q < 288 / 4; ++q) {
        v4f wv = wrow[q];
        v4f pv = prow[q];
        acc += pv[0] * wv[0];
        acc += pv[1] * wv[1];
        acc += pv[2] * wv[2];
        acc += pv[3] * wv[3];
    }
    out[((size_t)(b * OCH + o)) * TT + t0 + tt] = acc;
}

// ---------------------------------------------------------------------------
extern "C" void kernel_launch(void* const* d_in, const int* in_sizes, int n_in,
                              void* d_out, int out_size, void* d_ws, size_t ws_size,
                              hipStream_t stream) {
    const float* x      = (const float*)d_in[0];
    const float* weight = (const float*)d_in[1];
    const float* bias   = (const float*)d_in[2];
    float* out = (float*)d_out;

    // Workspace layout: xn (B*T*C f32) | topv (B*T*9 f32) | topi (B*T*9 i32)
    float* xn   = (float*)d_ws;
    float* topv = xn + (size_t)BB * TT * CCH;
    int*   topi = (int*)(topv + (size_t)BB * TT * KNN);

    normalize_kernel<<<(BB * TT) / 256, 256, 0, stream>>>(x, xn);
    simtopk_kernel<<<BB * (TT / 16), 32, 0, stream>>>(xn, topv, topi);
    gatherconv_kernel<<<BB * (TT / 16), 512, 0, stream>>>(x, weight, bias, topv, topi, out);
}